// Protein_Encoder_34342558499357
// MI455X (gfx1250) — compile-verified
//
#include <hip/hip_runtime.h>

typedef float v2f __attribute__((ext_vector_type(2)));
typedef float v8f __attribute__((ext_vector_type(8)));

#define HD 64

// ---------------- utility kernels ----------------

__global__ void zero_kernel(float* __restrict__ p, long n) {
  long i = (long)blockIdx.x * blockDim.x + threadIdx.x;
  long stride = (long)gridDim.x * blockDim.x;
  for (; i < n; i += stride) p[i] = 0.0f;
}

__global__ void degree_kernel(const int* __restrict__ dst, float* __restrict__ deg, int E) {
  int i = blockIdx.x * blockDim.x + threadIdx.x;
  int stride = gridDim.x * blockDim.x;
  for (; i < E; i += stride) atomicAdd(&deg[dst[i]], 1.0f);
}

__global__ void bcount_kernel(const int* __restrict__ batch, float* __restrict__ bcnt, int N) {
  int i = blockIdx.x * blockDim.x + threadIdx.x;
  int stride = gridDim.x * blockDim.x;
  for (; i < N; i += stride) atomicAdd(&bcnt[batch[i]], 1.0f);
}

__global__ void rinv_kernel(float* __restrict__ deg, int N) {
  int i = blockIdx.x * blockDim.x + threadIdx.x;
  int stride = gridDim.x * blockDim.x;
  for (; i < N; i += stride) deg[i] = 1.0f / fmaxf(deg[i], 1.0f);
}

// ---------------- WMMA fp32 GEMM: Y[N x 64] = A[N x K] @ W[K x 64] (+bias) ----------------
// blockDim = 128 (4 wave32s). Wave w computes the 16x16 tile at cols [16w,16w+16).
// A-matrix VGPR layout (16x4 f32): lanes 0-15 hold K=0,1; lanes 16-31 hold K=2,3.
// B-matrix (4x16): mirrored; C/D: 8 VGPRs, M = r + 8*(lane>=16).

__global__ void gemm_n64_wmma(const float* __restrict__ A, int K,
                              const float* __restrict__ W,
                              const float* __restrict__ bias,
                              float* __restrict__ Y, int Nrows) {
  const int lane = threadIdx.x & 31;
  const int wave = threadIdx.x >> 5;
  const int m0 = blockIdx.x * 16;
  const int n0 = wave * 16;
  const int lr = lane & 15;
  const int hi = lane >> 4;       // 0 for lanes 0-15, 1 for lanes 16-31
  const int kp = hi * 2;          // K sub-offset 0 or 2

  int row = m0 + lr;
  if (row >= Nrows) row = Nrows - 1;    // clamped load; stores guarded below
  const int n = n0 + lr;

  v8f c = {};
  const float* arow = A + (long)row * K;
  #pragma unroll 4
  for (int k0 = 0; k0 < K; k0 += 4) {
    v2f a, b;
    const float2 av = *(const float2*)(arow + k0 + kp);
    a.x = av.x; a.y = av.y;
    const float* wp = W + (long)(k0 + kp) * HD + n;
    b.x = wp[0];
    b.y = wp[HD];
    c = __builtin_amdgcn_wmma_f32_16x16x4_f32(
        /*neg_a=*/false, a, /*neg_b=*/false, b,
        /*c_mod=*/(short)0, c, /*reuse_a=*/false, /*reuse_b=*/false);
  }

  const float bv = bias ? bias[n] : 0.0f;
  #pragma unroll
  for (int r = 0; r < 8; ++r) {
    int orow = m0 + r + (hi << 3);
    if (orow < Nrows) Y[(long)orow * HD + n] = c[r] + bv;
  }
}

// ---------------- edge scatter: agg[dst] += F[src]  (64 floats/row) ----------------
// 4 threads per edge, 16 floats each; float4 loads, f32 atomics (resolve in L2).

__global__ void scatter_kernel(const float* __restrict__ F,
                               const int* __restrict__ src,
                               const int* __restrict__ dst,
                               float* __restrict__ agg, int E) {
  long i = (long)blockIdx.x * blockDim.x + threadIdx.x;
  long total = (long)E * 4;
  long stride = (long)gridDim.x * blockDim.x;
  for (; i < total; i += stride) {
    int e = (int)(i >> 2);
    int q = (int)(i & 3);
    const float4* fp = (const float4*)(F + (long)src[e] * HD + q * 16);
    float4 v0 = fp[0], v1 = fp[1], v2 = fp[2], v3 = fp[3];
    float* d = agg + (long)dst[e] * HD + q * 16;
    atomicAdd(d + 0,  v0.x); atomicAdd(d + 1,  v0.y);
    atomicAdd(d + 2,  v0.z); atomicAdd(d + 3,  v0.w);
    atomicAdd(d + 4,  v1.x); atomicAdd(d + 5,  v1.y);
    atomicAdd(d + 6,  v1.z); atomicAdd(d + 7,  v1.w);
    atomicAdd(d + 8,  v2.x); atomicAdd(d + 9,  v2.y);
    atomicAdd(d + 10, v2.z); atomicAdd(d + 11, v2.w);
    atomicAdd(d + 12, v3.x); atomicAdd(d + 13, v3.y);
    atomicAdd(d + 14, v3.z); atomicAdd(d + 15, v3.w);
  }
}

// ---------------- h_pre = agg*rinv + self ; accumulate BN sum / sumsq ----------------

__global__ void hpre_stats_kernel(const float* __restrict__ agg,
                                  const float* __restrict__ rinv,
                                  const float* __restrict__ hs,
                                  float* __restrict__ h,
                                  float* __restrict__ stats, int N) {
  const int f = threadIdx.x & 63;
  const int rpb = blockDim.x >> 6;
  int n = blockIdx.x * rpb + (threadIdx.x >> 6);
  const int stride = gridDim.x * rpb;
  float s = 0.0f, sq = 0.0f;
  for (; n < N; n += stride) {
    long idx = (long)n * HD + f;
    float v = agg[idx] * rinv[n] + hs[idx];
    h[idx] = v;
    s += v; sq += v * v;
  }
  atomicAdd(&stats[f], s);
  atomicAdd(&stats[64 + f], sq);
}

__global__ void bn_prep_kernel(const float* __restrict__ stats,
                               const float* __restrict__ gamma,
                               const float* __restrict__ beta,
                               float* __restrict__ ssh, float Ninv) {
  int f = threadIdx.x;
  if (f < 64) {
    float mu = stats[f] * Ninv;
    float var = stats[64 + f] * Ninv - mu * mu;   // biased variance (jnp.var)
    float rstd = rsqrtf(var + 1e-5f);
    float sc = rstd * gamma[f];
    ssh[f] = sc;
    ssh[64 + f] = beta[f] - mu * sc;
  }
}

__global__ void bn_apply_kernel(float* __restrict__ h, const float* __restrict__ ssh, long total) {
  long i = (long)blockIdx.x * blockDim.x + threadIdx.x;
  long stride = (long)gridDim.x * blockDim.x;
  for (; i < total; i += stride) {
    int f = (int)(i & 63);
    h[i] = fmaxf(h[i] * ssh[f] + ssh[64 + f], 0.0f);   // BN + ReLU
  }
}

// ---------------- h2 = agg2*rinv + hr ; pool[batch[n]] += h2 ----------------

__global__ void pool_kernel(const float* __restrict__ agg,
                            const float* __restrict__ rinv,
                            const float* __restrict__ hr,
                            const int* __restrict__ batch,
                            float* __restrict__ pool, long total) {
  long i = (long)blockIdx.x * blockDim.x + threadIdx.x;
  long stride = (long)gridDim.x * blockDim.x;
  for (; i < total; i += stride) {
    long n = i >> 6;
    int f = (int)(i & 63);
    float v = agg[i] * rinv[n] + hr[i];
    atomicAdd(&pool[(long)batch[n] * HD + f], v);
  }
}

__global__ void out_kernel(const float* __restrict__ pool,
                           const float* __restrict__ bcnt,
                           float* __restrict__ out, int BH) {
  int i = blockIdx.x * blockDim.x + threadIdx.x;
  if (i < BH) out[i] = pool[i] / fmaxf(bcnt[i >> 6], 1.0f);
}

// ---------------- driver ----------------

extern "C" void kernel_launch(void* const* d_in, const int* in_sizes, int n_in,
                              void* d_out, int out_size, void* d_ws, size_t ws_size,
                              hipStream_t stream) {
  (void)n_in; (void)ws_size;
  const float* x   = (const float*)d_in[0];
  const int*  eidx = (const int*)d_in[1];
  const int*  batch= (const int*)d_in[2];
  const float* W1l = (const float*)d_in[3];
  const float* b1  = (const float*)d_in[4];
  const float* W1r = (const float*)d_in[5];
  const float* g1  = (const float*)d_in[6];
  const float* be1 = (const float*)d_in[7];
  const float* W2l = (const float*)d_in[8];
  const float* b2  = (const float*)d_in[9];
  const float* W2r = (const float*)d_in[10];

  const int D = 128;
  const int N = in_sizes[0] / D;
  const int E = in_sizes[1] / 2;
  const int B = out_size / HD;
  const int* src = eidx;
  const int* dst = eidx + E;

  // workspace layout (floats); buffers reused across the two layers
  float* ws   = (float*)d_ws;
  const long NH = (long)N * HD;
  float* xl   = ws;            // x@W1l, then h@W2l
  float* hs   = xl + NH;       // x@W1r + b1, then h@W2r + b2
  float* agg  = hs + NH;       // edge accumulator (both layers)
  float* h    = agg + NH;      // h_pre, then BN+ReLU(h) in place
  float* rinv = h + NH;        // degree counts, then 1/max(deg,1)
  float* bcnt = rinv + N;      // 16
  float* stats= bcnt + 16;     // 128: per-feature sum, sumsq
  float* ssh  = stats + 128;   // 128: BN scale, shift
  float* pool = ssh + 128;     // B*64

  // init
  zero_kernel<<<2048, 256, 0, stream>>>(agg, NH);
  zero_kernel<<<256, 256, 0, stream>>>(rinv, N);
  zero_kernel<<<1, 256, 0, stream>>>(bcnt, 16 + 128 + 128 + (long)B * HD);

  degree_kernel<<<2048, 256, 0, stream>>>(dst, rinv, E);
  bcount_kernel<<<512, 256, 0, stream>>>(batch, bcnt, N);
  rinv_kernel<<<512, 256, 0, stream>>>(rinv, N);

  const int rowTiles = (N + 15) / 16;

  // layer 1: project first (D->H), then scatter on 64-wide rows (halves edge bytes)
  gemm_n64_wmma<<<rowTiles, 128, 0, stream>>>(x, D, W1l, nullptr, xl, N);
  gemm_n64_wmma<<<rowTiles, 128, 0, stream>>>(x, D, W1r, b1, hs, N);
  scatter_kernel<<<4096, 256, 0, stream>>>(xl, src, dst, agg, E);

  // BN (training-mode batch stats) + ReLU
  hpre_stats_kernel<<<512, 256, 0, stream>>>(agg, rinv, hs, h, stats, N);
  bn_prep_kernel<<<1, 64, 0, stream>>>(stats, g1, be1, ssh, 1.0f / (float)N);
  bn_apply_kernel<<<2048, 256, 0, stream>>>(h, ssh, NH);

  // layer 2
  gemm_n64_wmma<<<rowTiles, 128, 0, stream>>>(h, HD, W2l, nullptr, xl, N);
  gemm_n64_wmma<<<rowTiles, 128, 0, stream>>>(h, HD, W2r, b2, hs, N);
  zero_kernel<<<2048, 256, 0, stream>>>(agg, NH);
  scatter_kernel<<<4096, 256, 0, stream>>>(xl, src, dst, agg, E);

  // finalize + global mean pool
  pool_kernel<<<2048, 256, 0, stream>>>(agg, rinv, hs, batch, pool, NH);
  out_kernel<<<(B * HD + 255) / 256, 256, 0, stream>>>(pool, bcnt, (float*)d_out, B * HD);
}